// BANLayer1_32736240730177
// MI455X (gfx1250) — compile-verified
//
#include <hip/hip_runtime.h>
#include <math.h>

// ---------------- problem constants ----------------
#define BATCH     32
#define LV        1000
#define LQ        1200
#define VDIM      128
#define HID       512
#define NHEADS    8
#define HEAD_DIM  64
#define LEN_PRO   200
#define LP_PAD    208          // 13 * 16
#define MTILES_Q  13
#define MTILES_K  75           // 1200 / 16
#define NGROUPS   8            // 512 / 64
#define SCALE_INV 0.125f       // 1/sqrt(64)
#define BN_EPS    1e-5f

typedef __attribute__((ext_vector_type(2))) float v2f;
typedef __attribute__((ext_vector_type(8))) float v8f;

// =====================================================================
// Kernel 1: per-batch top-200 (stable, descending) + gather q_sel,
//           zero x_accum. One block (256 threads) per batch.
// =====================================================================
__global__ __launch_bounds__(256)
void topk_gather_kernel(const float* __restrict__ query,  // [B,LV,VDIM]
                        const float* __restrict__ att,    // [B,4,LV]
                        float* __restrict__ q_sel,        // [B,LP_PAD,VDIM]
                        float* __restrict__ out_idx,      // [B,LEN_PRO]
                        float* __restrict__ x_accum)      // [B,HID]
{
    const int b   = blockIdx.x;
    const int tid = threadIdx.x;

    __shared__ float sc[LV];
    __shared__ float rv[256];
    __shared__ int   ri[256];
    __shared__ int   bestIdx;

    const float* attb = att + (size_t)b * 4 * LV;
    for (int i = tid; i < LV; i += 256) {
        float m = attb[i];
        m = fmaxf(m, attb[LV     + i]);
        m = fmaxf(m, attb[2 * LV + i]);
        m = fmaxf(m, attb[3 * LV + i]);
        sc[i] = m;
    }
    for (int i = tid; i < HID; i += 256) x_accum[(size_t)b * HID + i] = 0.0f;
    __syncthreads();

    for (int t = 0; t < LEN_PRO; ++t) {
        float bv = -INFINITY; int bi = 0x7fffffff;
        for (int i = tid; i < LV; i += 256) {
            float v = sc[i];
            if (v > bv) { bv = v; bi = i; }   // i ascending per thread -> stable
        }
        rv[tid] = bv; ri[tid] = bi;
        __syncthreads();
        for (int s = 128; s > 0; s >>= 1) {
            if (tid < s) {
                float ov = rv[tid + s]; int oi = ri[tid + s];
                if (ov > rv[tid] || (ov == rv[tid] && oi < ri[tid])) { rv[tid] = ov; ri[tid] = oi; }
            }
            __syncthreads();
        }
        if (tid == 0) {
            bestIdx = ri[0];
            sc[ri[0]] = -INFINITY;
            out_idx[(size_t)b * LEN_PRO + t] = (float)ri[0];
        }
        __syncthreads();
        const int src = bestIdx;
        for (int k = tid; k < VDIM; k += 256)
            q_sel[((size_t)b * LP_PAD + t) * VDIM + k] =
                query[((size_t)b * LV + src) * VDIM + k];
        __syncthreads();
    }
    // zero pad rows 200..207
    for (int t = LEN_PRO; t < LP_PAD; ++t)
        for (int k = tid; k < VDIM; k += 256)
            q_sel[((size_t)b * LP_PAD + t) * VDIM + k] = 0.0f;
}

// =====================================================================
// Kernel 2: projection GEMM  Y[b,m,n] = (X[b,m,:] (+PE)) @ W + bias
//   One wave per 16x64 output tile, K=128 via 32 x v_wmma_f32_16x16x4_f32
//   per N-subtile (128 WMMAs per wave).
//   grid = (Mtiles, 8 n-groups, B), block = 32 (one wave)
// =====================================================================
__global__ __launch_bounds__(32)
void proj_gemm_kernel(const float* __restrict__ X,    // [B, Mrows, VDIM]
                      const float* __restrict__ W,    // [VDIM, HID]
                      const float* __restrict__ bias, // [HID]
                      float* __restrict__ Y,          // [B, Mrows, HID]
                      int Mrows, int apply_pe)
{
    const int mt   = blockIdx.x;
    const int ng   = blockIdx.y;
    const int b    = blockIdx.z;
    const int lane = threadIdx.x;
    const int lm   = lane & 15;   // m (A) / n (B,D) within tile
    const int lh   = lane >> 4;   // half select -> k = 2*lh + r

    const int m  = mt * 16 + lm;
    const int n0 = ng * 64;

    const float* Xrow = X + ((size_t)b * Mrows + m) * VDIM;

    v8f c0 = {}, c1 = {}, c2 = {}, c3 = {};

    const float neg_ln1e4_d = -9.2103403719761836f / 128.0f; // -log(10000)/VDIM
    const float mf = (float)m;

    for (int k0 = 0; k0 < VDIM; k0 += 4) {
        const int ka = k0 + 2 * lh;
        v2f a;
        a.x = Xrow[ka];
        a.y = Xrow[ka + 1];
        if (apply_pe) {
            // element ka: pair base = ka & ~1; even index -> sin, odd -> cos
            float ang0 = mf * __expf(neg_ln1e4_d * (float)(ka & ~1));
            a.x += (ka & 1) ? __cosf(ang0) : __sinf(ang0);
            const int kb = ka + 1;
            float ang1 = mf * __expf(neg_ln1e4_d * (float)(kb & ~1));
            a.y += (kb & 1) ? __cosf(ang1) : __sinf(ang1);
        }
        const float* Wk0 = W + (size_t)ka * HID;
        const float* Wk1 = W + (size_t)(ka + 1) * HID;
        v2f bb;
        bb.x = Wk0[n0 + 0 * 16 + lm]; bb.y = Wk1[n0 + 0 * 16 + lm];
        c0 = __builtin_amdgcn_wmma_f32_16x16x4_f32(false, a, false, bb, (short)0, c0, false, false);
        bb.x = Wk0[n0 + 1 * 16 + lm]; bb.y = Wk1[n0 + 1 * 16 + lm];
        c1 = __builtin_amdgcn_wmma_f32_16x16x4_f32(false, a, false, bb, (short)0, c1, false, false);
        bb.x = Wk0[n0 + 2 * 16 + lm]; bb.y = Wk1[n0 + 2 * 16 + lm];
        c2 = __builtin_amdgcn_wmma_f32_16x16x4_f32(false, a, false, bb, (short)0, c2, false, false);
        bb.x = Wk0[n0 + 3 * 16 + lm]; bb.y = Wk1[n0 + 3 * 16 + lm];
        c3 = __builtin_amdgcn_wmma_f32_16x16x4_f32(false, a, false, bb, (short)0, c3, false, false);
    }

    // D layout: n = lm, m = r + 8*lh
    v8f cs[4] = {c0, c1, c2, c3};
#pragma unroll
    for (int nt = 0; nt < 4; ++nt) {
        const int n = n0 + nt * 16 + lm;
        const float bsn = bias[n];
#pragma unroll
        for (int r = 0; r < 8; ++r) {
            const int mr = mt * 16 + r + 8 * lh;
            Y[((size_t)b * Mrows + mr) * HID + n] = cs[nt][r] + bsn;
        }
    }
}

// =====================================================================
// Kernel 3: fused attention.
//   block = 256 threads = 8 waves = 8 heads; grid = (13 v-tiles, B)
//   Per q-tile: S = Q.K^T * 1/8 (16 WMMAs), head-sum -> attn_avg,
//   then t += S @ V2 (16 WMMAs). After q-loop: x += sum_v V1*t (masked).
// =====================================================================
__global__ __launch_bounds__(256)
void attn_kernel(const float* __restrict__ Qp,       // [B,LP_PAD,HID]
                 const float* __restrict__ Kp,       // [B,LQ,HID]
                 const float* __restrict__ V1,       // [B,LP_PAD,HID]
                 const float* __restrict__ V2,       // [B,LQ,HID]
                 float* __restrict__ out_attn,       // [B,LEN_PRO,LQ]
                 float* __restrict__ x_accum)        // [B,HID]
{
    const int vt   = blockIdx.x;
    const int b    = blockIdx.y;
    const int tid  = threadIdx.x;
    const int h    = tid >> 5;      // wave id = head
    const int lane = tid & 31;
    const int lm   = lane & 15;
    const int lh   = lane >> 4;

    __shared__ float Slds[NHEADS * 16 * 16];   // per-head scaled S tile

    // ---- preload Q A-fragments for this (b, v-tile, head): 16 k-subtiles ----
    const int vrowA = vt * 16 + lm;
    const float* Qrow = Qp + ((size_t)b * LP_PAD + vrowA) * HID + h * HEAD_DIM;
    v2f qa[16];
#pragma unroll
    for (int kk = 0; kk < 16; ++kk) {
        const int ka = kk * 4 + 2 * lh;
        qa[kk].x = Qrow[ka];
        qa[kk].y = Qrow[ka + 1];
    }

    v8f t0 = {}, t1 = {}, t2 = {}, t3 = {};   // t tile: 16 v-rows x 64 d

    const int   avloc = tid >> 4;             // cooperative attn_avg mapping
    const int   aqloc = tid & 15;
    const int   avrow = vt * 16 + avloc;
    float* Sme = &Slds[h * 256];

    for (int q0 = 0; q0 < LQ; q0 += 16) {
        // ---- S = Q . K^T ----
        v8f s = {};
        const float* Krow = Kp + ((size_t)b * LQ + q0 + lm) * HID + h * HEAD_DIM;
#pragma unroll
        for (int kk = 0; kk < 16; ++kk) {
            const int ka = kk * 4 + 2 * lh;
            v2f kb;
            kb.x = Krow[ka];
            kb.y = Krow[ka + 1];
            s = __builtin_amdgcn_wmma_f32_16x16x4_f32(false, qa[kk], false, kb, (short)0, s, false, false);
        }
        // scale and stage to LDS (element (m = r+8*lh, n = lm))
#pragma unroll
        for (int r = 0; r < 8; ++r)
            Sme[(r + 8 * lh) * 16 + lm] = s[r] * SCALE_INV;
        __syncthreads();

        // ---- head-average write (attn_avg = sum_h attn / 8) ----
        if (avrow < LEN_PRO) {
            float acc = 0.f;
#pragma unroll
            for (int hh = 0; hh < NHEADS; ++hh)
                acc += Slds[hh * 256 + avloc * 16 + aqloc];
            out_attn[((size_t)b * LEN_PRO + avrow) * LQ + q0 + aqloc] = acc * (1.0f / NHEADS);
        }

        // ---- t += S @ V2 (A = S from LDS, transposed layout for free) ----
#pragma unroll
        for (int kk2 = 0; kk2 < 4; ++kk2) {
            const int kq = kk2 * 4 + 2 * lh;     // q-local k index
            v2f a2;
            a2.x = Sme[lm * 16 + kq];
            a2.y = Sme[lm * 16 + kq + 1];
            const float* V2r0 = V2 + ((size_t)b * LQ + q0 + kq)     * HID + h * HEAD_DIM + lm;
            const float* V2r1 = V2 + ((size_t)b * LQ + q0 + kq + 1) * HID + h * HEAD_DIM + lm;
            v2f bb;
            bb.x = V2r0[0 * 16]; bb.y = V2r1[0 * 16];
            t0 = __builtin_amdgcn_wmma_f32_16x16x4_f32(false, a2, false, bb, (short)0, t0, false, false);
            bb.x = V2r0[1 * 16]; bb.y = V2r1[1 * 16];
            t1 = __builtin_amdgcn_wmma_f32_16x16x4_f32(false, a2, false, bb, (short)0, t1, false, false);
            bb.x = V2r0[2 * 16]; bb.y = V2r1[2 * 16];
            t2 = __builtin_amdgcn_wmma_f32_16x16x4_f32(false, a2, false, bb, (short)0, t2, false, false);
            bb.x = V2r0[3 * 16]; bb.y = V2r1[3 * 16];
            t3 = __builtin_amdgcn_wmma_f32_16x16x4_f32(false, a2, false, bb, (short)0, t3, false, false);
        }
        __syncthreads();   // protect Slds before next iteration overwrites
    }

    // ---- x += sum over (valid) v of V1 * t ----
    v8f ts[4] = {t0, t1, t2, t3};
#pragma unroll
    for (int nt = 0; nt < 4; ++nt) {
        const int d = h * HEAD_DIM + nt * 16 + lm;
        float partial = 0.f;
#pragma unroll
        for (int r = 0; r < 8; ++r) {
            const int vr = vt * 16 + r + 8 * lh;
            if (vr < LEN_PRO)
                partial += V1[((size_t)b * LP_PAD + vr) * HID + d] * ts[nt][r];
        }
        atomicAdd(&x_accum[(size_t)b * HID + d], partial);
    }
}

// =====================================================================
// Kernel 4: batchnorm finalize
// =====================================================================
__global__ __launch_bounds__(256)
void bn_kernel(const float* __restrict__ x_accum,
               const float* __restrict__ mean, const float* __restrict__ var,
               const float* __restrict__ gamma, const float* __restrict__ beta,
               float* __restrict__ out_x)
{
    const int i = blockIdx.x * 256 + threadIdx.x;
    if (i < BATCH * HID) {
        const int c = i & (HID - 1);
        out_x[i] = (x_accum[i] - mean[c]) * rsqrtf(var[c] + BN_EPS) * gamma[c] + beta[c];
    }
}

// =====================================================================
extern "C" void kernel_launch(void* const* d_in, const int* in_sizes, int n_in,
                              void* d_out, int out_size, void* d_ws, size_t ws_size,
                              hipStream_t stream)
{
    const float* query = (const float*)d_in[0];
    const float* key   = (const float*)d_in[1];
    const float* att   = (const float*)d_in[2];
    const float* Wq    = (const float*)d_in[3];
    const float* bq    = (const float*)d_in[4];
    const float* Wk    = (const float*)d_in[5];
    const float* bk    = (const float*)d_in[6];
    const float* Wv1   = (const float*)d_in[7];
    const float* bv1   = (const float*)d_in[8];
    const float* Wv2   = (const float*)d_in[9];
    const float* bv2   = (const float*)d_in[10];
    const float* gamma = (const float*)d_in[11];
    const float* beta  = (const float*)d_in[12];
    const float* mean  = (const float*)d_in[13];
    const float* var   = (const float*)d_in[14];

    float* ws = (float*)d_ws;
    size_t off = 0;
    float* q_sel   = ws + off; off += (size_t)BATCH * LP_PAD * VDIM;
    float* Qp      = ws + off; off += (size_t)BATCH * LP_PAD * HID;
    float* V1p     = ws + off; off += (size_t)BATCH * LP_PAD * HID;
    float* Kp      = ws + off; off += (size_t)BATCH * LQ * HID;
    float* V2p     = ws + off; off += (size_t)BATCH * LQ * HID;
    float* x_accum = ws + off; off += (size_t)BATCH * HID;

    float* out_x    = (float*)d_out;                 // [B,HID]
    float* out_idx  = out_x   + (size_t)BATCH * HID; // [B,LEN_PRO] (values as float)
    float* out_attn = out_idx + (size_t)BATCH * LEN_PRO; // [B,LEN_PRO,LQ]

    topk_gather_kernel<<<BATCH, 256, 0, stream>>>(query, att, q_sel, out_idx, x_accum);

    proj_gemm_kernel<<<dim3(MTILES_Q, NGROUPS, BATCH), 32, 0, stream>>>(
        q_sel, Wq,  bq,  Qp,  LP_PAD, /*pe=*/1);
    proj_gemm_kernel<<<dim3(MTILES_Q, NGROUPS, BATCH), 32, 0, stream>>>(
        q_sel, Wv1, bv1, V1p, LP_PAD, /*pe=*/0);
    proj_gemm_kernel<<<dim3(MTILES_K, NGROUPS, BATCH), 32, 0, stream>>>(
        key,   Wk,  bk,  Kp,  LQ,     /*pe=*/1);
    proj_gemm_kernel<<<dim3(MTILES_K, NGROUPS, BATCH), 32, 0, stream>>>(
        key,   Wv2, bv2, V2p, LQ,     /*pe=*/0);

    attn_kernel<<<dim3(MTILES_Q, BATCH), 256, 0, stream>>>(
        Qp, Kp, V1p, V2p, out_attn, x_accum);

    bn_kernel<<<(BATCH * HID + 255) / 256, 256, 0, stream>>>(
        x_accum, mean, var, gamma, beta, out_x);
}